// GATLayer_43456479101350
// MI455X (gfx1250) — compile-verified
//
#include <hip/hip_runtime.h>
#include <hip/hip_bf16.h>

typedef __attribute__((ext_vector_type(2))) float v2f;
typedef __attribute__((ext_vector_type(8))) float v8f;

#define N_NODES  100000
#define N_FEAT   256
#define N_HIDDEN 64
#define LEAKY    0.05f

// ---------------------------------------------------------------------------
// Kernel 0: zero the accumulation targets (out matrix + row_sum).
// Must run every launch: harness poisons once, we atomically accumulate.
// ---------------------------------------------------------------------------
__global__ void k0_zero(float* __restrict__ out, float* __restrict__ row_sum,
                        int n_out, int n_rs) {
    int i = blockIdx.x * blockDim.x + threadIdx.x;
    if (i < n_out) out[i] = 0.0f;
    if (i < n_rs)  row_sum[i] = 0.0f;
}

// ---------------------------------------------------------------------------
// Kernel 1: h = x @ W_fc via V_WMMA_F32_16X16X4_F32 (exact fp32 tensor path),
// plus fused s = h @ a_w[:64], t = h @ a_w[64:] via wave32 shuffle reduction.
// Block = 128 threads = 4 waves; each wave owns a 16-row tile, block owns 64
// rows x 64 cols. W_fc staged transposed into LDS so each B fragment is a
// contiguous 8-byte LDS load.
// ---------------------------------------------------------------------------
__global__ void __launch_bounds__(128)
k1_gemm(const float* __restrict__ x, const float* __restrict__ W,
        const float* __restrict__ a_w, float* __restrict__ h,
        float* __restrict__ s_out, float* __restrict__ t_out) {
    __shared__ float Wt[N_HIDDEN * N_FEAT];  // Wt[n][k], 64 KB

    const int tid = threadIdx.x;
    // Stage + transpose W (memory is [k][n] row-major) into Wt[n][k].
    for (int idx = tid; idx < N_FEAT * N_HIDDEN; idx += 128) {
        int k = idx >> 6;      // / 64
        int n = idx & 63;
        Wt[n * N_FEAT + k] = W[idx];
    }
    __syncthreads();

    const int wave = tid >> 5;
    const int lane = tid & 31;
    const int nl   = lane & 15;   // column-within-tile / A-row select
    const int hi   = lane >> 4;   // half-wave: selects K pair for A/B frags

    const int rowbase = blockIdx.x * 64 + wave * 16;
    int rowA = rowbase + nl;
    if (rowA >= N_NODES) rowA = N_NODES - 1;          // clamp loads, predicate stores
    const float* __restrict__ xr = x + (size_t)rowA * N_FEAT;

    v8f acc[4];
#pragma unroll
    for (int c = 0; c < 4; ++c) { v8f z = {}; acc[c] = z; }

    // K loop: one A frag load (b64), four B frag LDS loads (b64), four WMMAs.
#pragma unroll 4
    for (int k0 = 0; k0 < N_FEAT; k0 += 4) {
        v2f a;
        a.x = xr[k0 + 2 * hi];
        a.y = xr[k0 + 2 * hi + 1];
#pragma unroll
        for (int c = 0; c < 4; ++c) {
            const float* wp = &Wt[(c * 16 + nl) * N_FEAT + k0 + 2 * hi];
            v2f b;
            b.x = wp[0];
            b.y = wp[1];
            acc[c] = __builtin_amdgcn_wmma_f32_16x16x4_f32(
                false, a, false, b, (short)0, acc[c], false, false);
        }
    }

    // Per-lane a_w slices for the fused s/t projections.
    float aws[4], awt[4];
#pragma unroll
    for (int c = 0; c < 4; ++c) {
        aws[c] = a_w[c * 16 + nl];
        awt[c] = a_w[N_HIDDEN + c * 16 + nl];
    }

    // C/D layout: VGPR j -> row M = j + 8*hi, column N = nl + 16*c.
#pragma unroll
    for (int j = 0; j < 8; ++j) {
        const int r = rowbase + 8 * hi + j;
        float ps = 0.0f, pt = 0.0f;
#pragma unroll
        for (int c = 0; c < 4; ++c) {
            float v = acc[c][j];
            ps += v * aws[c];
            pt += v * awt[c];
            if (r < N_NODES)
                h[(size_t)r * N_HIDDEN + c * 16 + nl] = v;
        }
        // Reduce across the 16 lanes of this half-wave (all different columns
        // of the same row): xor-shuffle tree, wave32.
#pragma unroll
        for (int m = 1; m < 16; m <<= 1) {
            ps += __shfl_xor(ps, m, 32);
            pt += __shfl_xor(pt, m, 32);
        }
        if (nl == 0 && r < N_NODES) {
            s_out[r] = ps;
            t_out[r] = pt;
        }
    }
}

// ---------------------------------------------------------------------------
// Kernel 2: per-edge raw attention + softmax denominator.
// s/t/row_sum are ~400 KB each -> L2 resident; atomics resolve in L2.
// ---------------------------------------------------------------------------
__global__ void k2_att(const int* __restrict__ ei, const float* __restrict__ s,
                       const float* __restrict__ t, const float* __restrict__ a_b,
                       float* __restrict__ att, float* __restrict__ row_sum, int E) {
    int e = blockIdx.x * blockDim.x + threadIdx.x;
    if (e >= E) return;
    int src = ei[e];
    int dst = ei[E + e];
    float v = s[src] + t[dst] + a_b[0];
    v = (v >= 0.0f) ? v : LEAKY * v;
    float a = expf(v);
    att[e] = a;
    atomicAdd(&row_sum[src], a);
}

// ---------------------------------------------------------------------------
// Kernel 3: alpha = att / row_sum[src]; out[src] += alpha * h[dst].
// One wave per edge: 32 lanes cover the 64 hidden channels (2 floats each),
// giving coalesced 128B reads of h[dst] and 64 f32 scatter-atomics into
// out[src] — both L2-resident (h + out = 51 MB << 192 MB L2).
// ---------------------------------------------------------------------------
__global__ void __launch_bounds__(256)
k3_aggr(const int* __restrict__ ei, const float* __restrict__ h,
        const float* __restrict__ att, const float* __restrict__ row_sum,
        float* __restrict__ out, float* __restrict__ alpha_out, int E) {
    const int lane   = threadIdx.x & 31;
    const int gwarp  = (blockIdx.x * blockDim.x + threadIdx.x) >> 5;
    const int nwarp  = (gridDim.x * blockDim.x) >> 5;

    for (int e = gwarp; e < E; e += nwarp) {
        int src = ei[e];
        int dst = ei[E + e];
        float alpha = att[e] / row_sum[src];
        if (lane == 0) alpha_out[e] = alpha;

        const float* hd = h + (size_t)dst * N_HIDDEN;
        float* od = out + (size_t)src * N_HIDDEN;
        float h0 = hd[lane];
        float h1 = hd[lane + 32];
        atomicAdd(&od[lane],      alpha * h0);
        atomicAdd(&od[lane + 32], alpha * h1);
    }
}

// ---------------------------------------------------------------------------
extern "C" void kernel_launch(void* const* d_in, const int* in_sizes, int n_in,
                              void* d_out, int out_size, void* d_ws, size_t ws_size,
                              hipStream_t stream) {
    const float* x   = (const float*)d_in[0];
    const int*   ei  = (const int*)d_in[1];     // edge_index [2][E]
    const float* W   = (const float*)d_in[2];   // [256][64]
    const float* a_w = (const float*)d_in[3];   // [128]
    const float* a_b = (const float*)d_in[4];   // [1]

    float* out       = (float*)d_out;                         // [N, 64]
    float* alpha_out = out + (size_t)N_NODES * N_HIDDEN;      // [E]

    const int E = in_sizes[1] / 2;

    // Workspace carve-up (~40 MB total).
    char*  ws      = (char*)d_ws;
    float* h       = (float*)ws; ws += (size_t)N_NODES * N_HIDDEN * sizeof(float);
    float* s       = (float*)ws; ws += (size_t)N_NODES * sizeof(float);
    float* t       = (float*)ws; ws += (size_t)N_NODES * sizeof(float);
    float* row_sum = (float*)ws; ws += (size_t)N_NODES * sizeof(float);
    float* att     = (float*)ws;

    // 0) zero accumulation targets
    {
        const int n_out = N_NODES * N_HIDDEN;
        const int blk = 256;
        k0_zero<<<(n_out + blk - 1) / blk, blk, 0, stream>>>(out, row_sum,
                                                             n_out, N_NODES);
    }
    // 1) WMMA GEMM + fused s/t projections
    {
        const int grd = (N_NODES + 63) / 64;
        k1_gemm<<<grd, 128, 0, stream>>>(x, W, a_w, h, s, t);
    }
    // 2) edge attention + softmax denominator
    {
        const int blk = 256;
        k2_att<<<(E + blk - 1) / blk, blk, 0, stream>>>(ei, s, t, a_b, att,
                                                        row_sum, E);
    }
    // 3) normalize + aggregate
    {
        k3_aggr<<<8192, 256, 0, stream>>>(ei, h, att, row_sum, out, alpha_out, E);
    }
}